// rpn_35854386987658
// MI455X (gfx1250) — compile-verified
//
#include <hip/hip_runtime.h>

typedef float v2f __attribute__((ext_vector_type(2)));
typedef float v4f __attribute__((ext_vector_type(4)));
typedef float v8f __attribute__((ext_vector_type(8)));

#define MPIX   15200          // B*H*W = 1*100*152
#define KDIM   1024
#define NPAD   64             // 18 cls + 36 bbox + 10 pad
#define NCLS   18
#define NBB    36
#define NANCH  136800         // MPIX * 9
#define KTOP   1000
#define NEGV   (-1.0e30f)
#define IOUT   0.7f

// ---------------------------------------------------------------------------
// Kernel 1: pack w_cls (1024x18) and w_bbox (1024x36) into Wpad (1024x64),
// biases into bpad[64]. Padding columns are zero.
// ---------------------------------------------------------------------------
__global__ void pack_weights_kernel(const float* __restrict__ w_cls,
                                    const float* __restrict__ b_cls,
                                    const float* __restrict__ w_bbox,
                                    const float* __restrict__ b_bbox,
                                    float* __restrict__ Wpad,
                                    float* __restrict__ bpad) {
  int t = blockIdx.x * blockDim.x + threadIdx.x;
  if (t < KDIM * NPAD) {
    int row = t >> 6, col = t & 63;
    float v = 0.f;
    if (col < NCLS)            v = w_cls[row * NCLS + col];
    else if (col < NCLS + NBB) v = w_bbox[row * NBB + (col - NCLS)];
    Wpad[t] = v;
  }
  if (t < NPAD) {
    float v = 0.f;
    if (t < NCLS)            v = b_cls[t];
    else if (t < NCLS + NBB) v = b_bbox[t - NCLS];
    bpad[t] = v;
  }
}

// ---------------------------------------------------------------------------
// Kernel 2: WMMA GEMM  sc[15200x64] = feat[15200x1024] * Wpad[1024x64] + bpad
// One block = 128 threads = 4 wave32; wave w handles N-tile w; block handles
// one 16-row M-tile, so the 4 waves share feat rows through L0 and feat is
// read from HBM exactly once (62 MB / 23.3 TB/s ~ 2.7 us floor).
// f32 WMMA: V_WMMA_F32_16X16X4_F32, K stepped by 4, 256 MACs per tile.
// A-frag (16x4 f32): lane = {khalf,row16}; VGPR0 = K(2*khalf), VGPR1 = K+1.
// C layout: VGPR r -> M = r + 8*khalf, N = row16.
// ---------------------------------------------------------------------------
__global__ __launch_bounds__(128) void gemm_wmma_kernel(
    const float* __restrict__ feat,
    const float* __restrict__ Wpad,
    const float* __restrict__ bpad,
    float* __restrict__ sc) {
  const int mtile = blockIdx.x;           // 0..949
  const int ntile = threadIdx.x >> 5;     // 0..3 (wave id)
  const int lane  = threadIdx.x & 31;
  const int row16 = lane & 15;
  const int khalf = lane >> 4;            // 0 or 1

  const float* aptr = feat + (size_t)(mtile * 16 + row16) * KDIM + 2 * khalf;
  const float* bcol = Wpad + ntile * 16 + row16;  // N column = row16

  v8f acc = {};
  for (int k = 0; k < KDIM; k += 4) {
    if ((k & 63) == 0) __builtin_prefetch(aptr + k + 128, 0, 3);
    v2f a = *(const v2f*)(aptr + k);      // {K = k+2*khalf, K+1}
    v2f b;
    b.x = bcol[(k + 2 * khalf)     * NPAD];
    b.y = bcol[(k + 2 * khalf + 1) * NPAD];
    acc = __builtin_amdgcn_wmma_f32_16x16x4_f32(
        /*neg_a=*/false, a, /*neg_b=*/false, b,
        /*c_mod=*/(short)0, acc, /*reuse_a=*/false, /*reuse_b=*/false);
  }

  const int outcol = ntile * 16 + row16;
  const float bias = bpad[outcol];
#pragma unroll
  for (int r = 0; r < 8; ++r) {
    const int outrow = mtile * 16 + r + 8 * khalf;
    sc[(size_t)outrow * NPAD + outcol] = acc[r] + bias;
  }
}

// ---------------------------------------------------------------------------
// Kernel 3: per-anchor softmax / argmax / bbox decode+clip.
// Anchor a = m*9 + j. cls pair j -> cls_prob channels (2j,2j+1) and cls_pred.
// fg score = cls_prob channel 9+j = softmax pair (9+j)>>1, elem (9+j)&1
// (exact replica of the reference's reshape semantics).
// Boxes stored packed as float4 (b128) for the NMS inner loop.
// ---------------------------------------------------------------------------
__global__ void decode_kernel(const float* __restrict__ sc,
                              const float* __restrict__ anchors,
                              const float* __restrict__ img_sz,
                              float* __restrict__ o_cls_prob,
                              float* __restrict__ o_cls_pred,
                              float* __restrict__ o_bbox,
                              v4f* __restrict__ boxes4,
                              float* __restrict__ bsc) {
  int a = blockIdx.x * blockDim.x + threadIdx.x;
  if (a >= NANCH) return;
  int m = a / 9, j = a - m * 9;
  const float* row = sc + (size_t)m * NPAD;

  // cls softmax pair j  -> cls_prob / cls_pred outputs
  float s0 = row[2 * j], s1 = row[2 * j + 1];
  float mx = fmaxf(s0, s1);
  float e0 = __expf(s0 - mx), e1 = __expf(s1 - mx);
  float inv = 1.f / (e0 + e1);
  o_cls_prob[m * NCLS + 2 * j]     = e0 * inv;
  o_cls_prob[m * NCLS + 2 * j + 1] = e1 * inv;
  o_cls_pred[a] = (s1 > s0) ? 1.f : 0.f;

  // foreground score: channel 9+j of cls_prob
  int ch = 9 + j, pr = ch >> 1;
  float f0 = row[2 * pr], f1 = row[2 * pr + 1];
  float fmx = fmaxf(f0, f1);
  float g0 = __expf(f0 - fmx), g1 = __expf(f1 - fmx);
  float gin = 1.f / (g0 + g1);
  float fg = (ch & 1) ? g1 * gin : g0 * gin;

  // bbox deltas
  float tx = row[NCLS + 4 * j + 0];
  float ty = row[NCLS + 4 * j + 1];
  float tw = row[NCLS + 4 * j + 2];
  float th = row[NCLS + 4 * j + 3];
  o_bbox[(size_t)a * 4 + 0] = tx;
  o_bbox[(size_t)a * 4 + 1] = ty;
  o_bbox[(size_t)a * 4 + 2] = tw;
  o_bbox[(size_t)a * 4 + 3] = th;

  // convert_and_clip
  float a0 = anchors[(size_t)a * 4 + 0], a1 = anchors[(size_t)a * 4 + 1];
  float a2 = anchors[(size_t)a * 4 + 2], a3 = anchors[(size_t)a * 4 + 3];
  float w = a3 - a1 + 1.f, h = a2 - a0 + 1.f;
  float x = a0 + 0.5f * h, y = a1 + 0.5f * w;
  float xp = tx * h + x, yp = ty * w + y;
  float wp = __expf(tw) * w, hp = __expf(th) * h;
  float i0 = img_sz[0], i1 = img_sz[1];
  v4f bx;
  bx.x = fmaxf(xp - 0.5f * hp, 0.f);
  bx.y = fmaxf(yp - 0.5f * wp, 0.f);
  bx.z = fminf(xp + 0.5f * hp, i0);
  bx.w = fminf(yp + 0.5f * wp, i1);
  boxes4[a] = bx;                    // single b128 store
  bsc[a] = fg;
}

// ---------------------------------------------------------------------------
// Kernel 4: greedy NMS, 1000 sequential rounds, single 1024-thread workgroup.
// Working set (~2.7 MB) is L2-resident. Boxes read as one b128 per element;
// area recomputed in VALU (identical formula to reference). LDS tree argmax
// with first-index tie-break (matches jnp.argmax).
// ---------------------------------------------------------------------------
__global__ __launch_bounds__(1024) void nms_kernel(
    const v4f* __restrict__ boxes4,
    float* __restrict__ bsc,
    float* __restrict__ o_prop, float* __restrict__ o_idx) {
  __shared__ float sVal[1024];
  __shared__ int   sIdx[1024];
  __shared__ float sBox[5];
  __shared__ int   sValid;
  const int tid = threadIdx.x;

  for (int t = 0; t < KTOP; ++t) {
    // local argmax over strided elements
    float best = -3.0e38f;
    int bi = 0x7fffffff;
    for (int j = tid; j < NANCH; j += 1024) {
      float v = bsc[j];
      if (v > best || (v == best && j < bi)) { best = v; bi = j; }
    }
    sVal[tid] = best; sIdx[tid] = bi;
    __syncthreads();
    // tree reduce, lowest-index wins ties
    for (int s = 512; s > 0; s >>= 1) {
      if (tid < s) {
        float v = sVal[tid + s]; int i2 = sIdx[tid + s];
        if (v > sVal[tid] || (v == sVal[tid] && i2 < sIdx[tid])) {
          sVal[tid] = v; sIdx[tid] = i2;
        }
      }
      __syncthreads();
    }
    if (tid == 0) {
      int i = sIdx[0];
      int valid = (sVal[0] > NEGV * 0.5f) ? 1 : 0;
      sValid = valid;
      v4f b = boxes4[i];
      sBox[0] = b.x; sBox[1] = b.y; sBox[2] = b.z; sBox[3] = b.w;
      sBox[4] = (b.z - b.x) * (b.w - b.y);      // area[i]
      o_idx[t] = (float)i;
      o_prop[4 * t + 0] = valid ? b.x : 0.f;
      o_prop[4 * t + 1] = valid ? b.y : 0.f;
      o_prop[4 * t + 2] = valid ? b.z : 0.f;
      o_prop[4 * t + 3] = valid ? b.w : 0.f;
      bsc[i] = NEGV;                            // always knocked out (s.at[i].set)
    }
    __syncthreads();
    if (sValid) {
      float X0 = sBox[0], Y0 = sBox[1], X1 = sBox[2], Y1 = sBox[3], AR = sBox[4];
      for (int j = tid; j < NANCH; j += 1024) {
        v4f b = boxes4[j];                      // one b128 load
        float iw = fmaxf(fminf(b.z, X1) - fmaxf(b.x, X0), 0.f);
        float ih = fmaxf(fminf(b.w, Y1) - fmaxf(b.y, Y0), 0.f);
        float inter = iw * ih;
        float area_j = (b.z - b.x) * (b.w - b.y);
        float iou = inter / (area_j + AR - inter + 1e-9f);
        if (iou >= IOUT) bsc[j] = NEGV;
      }
    }
    __threadfence();
    __syncthreads();
  }
}

// ---------------------------------------------------------------------------
extern "C" void kernel_launch(void* const* d_in, const int* in_sizes, int n_in,
                              void* d_out, int out_size, void* d_ws, size_t ws_size,
                              hipStream_t stream) {
  const float* feat    = (const float*)d_in[0];
  const float* anchors = (const float*)d_in[1];
  const float* img_sz  = (const float*)d_in[2];
  const float* w_cls   = (const float*)d_in[3];
  const float* b_cls   = (const float*)d_in[4];
  const float* w_bbox  = (const float*)d_in[5];
  const float* b_bbox  = (const float*)d_in[6];
  (void)in_sizes; (void)n_in; (void)out_size; (void)ws_size;

  // workspace layout (floats); boxes4 aligned to 16B by construction
  float* Wpad   = (float*)d_ws;                  // 1024*64
  float* bpad   = Wpad + KDIM * NPAD;            // 64
  float* sc     = bpad + NPAD;                   // 15200*64
  v4f*   boxes4 = (v4f*)(sc + (size_t)MPIX * NPAD);  // 136800 float4
  float* bsc    = (float*)(boxes4 + NANCH);      // 136800

  // output layout (floats), reference tuple order
  float* o        = (float*)d_out;
  float* o_prop   = o;                                   // 1000*4
  float* o_idx    = o_prop + KTOP * 4;                   // 1000
  float* o_clsp   = o_idx + KTOP;                        // 15200*18
  float* o_clspr  = o_clsp + (size_t)MPIX * NCLS;        // 136800
  float* o_bbox   = o_clspr + NANCH;                     // 136800*4

  pack_weights_kernel<<<(KDIM * NPAD + 255) / 256, 256, 0, stream>>>(
      w_cls, b_cls, w_bbox, b_bbox, Wpad, bpad);

  gemm_wmma_kernel<<<MPIX / 16, 128, 0, stream>>>(feat, Wpad, bpad, sc);

  decode_kernel<<<(NANCH + 255) / 256, 256, 0, stream>>>(
      sc, anchors, img_sz, o_clsp, o_clspr, o_bbox, boxes4, bsc);

  nms_kernel<<<1, 1024, 0, stream>>>(boxes4, bsc, o_prop, o_idx);
}